// LSTMClassifier_14096082665835
// MI455X (gfx1250) — compile-verified
//
#include <hip/hip_runtime.h>
#include <math.h>

// LSTM classifier: B=64, T=256, E=H=512, 4H=2048, O=5, fp32.
// Matrix path: V_WMMA_F32_16X16X4_F32 (fp32 in, fp32 accumulate) == reference
// precision. GEMM1 is LDS-tiled for data reuse; the recurrence keeps one wave
// per 16x16 batch-hidden patch with 4 gate accumulators (lane-local cell
// update, no LDS), h double-buffered across the 256 sequential steps.

typedef __attribute__((ext_vector_type(2))) float v2f;
typedef __attribute__((ext_vector_type(8))) float v8f;

#define B_   64
#define T_   256
#define E_   512
#define H_   512
#define G4_  2048
#define O_   5
#define PAD_ 1

#define MBLK   64          // rows per block in GEMM1 (4 m-subtiles of 16)
#define KCHUNK 64          // K chunk staged in LDS
#define APAD   4           // row pad -> 68-float stride: 16B-aligned b128 rows,
                           // 4-bank stride across the 16 fragment rows

__device__ __forceinline__ float sigmf(float v) {
    return 1.0f / (1.0f + expf(-v));
}

__device__ __forceinline__ v8f wmma_f32(v2f a, v2f b, v8f c) {
    // 8 args: (neg_a, A, neg_b, B, c_mod, C, reuse_a, reuse_b)
    return __builtin_amdgcn_wmma_f32_16x16x4_f32(false, a, false, b,
                                                 (short)0, c, false, false);
}

// ---------------------------------------------------------------------------
// lengths[b] = #non-pad tokens in row b
// ---------------------------------------------------------------------------
__global__ void lengths_kernel(const int* __restrict__ x, int* __restrict__ len) {
    int b = threadIdx.x;
    if (b < B_) {
        int c = 0;
        for (int t = 0; t < T_; ++t) c += (x[b * T_ + t] != PAD_) ? 1 : 0;
        len[b] = c;
    }
}

// ---------------------------------------------------------------------------
// xproj[t][b][g] = embed[x[b,t]] . wi_w[g] + wi_b[g]
//
// Hierarchical fp32 WMMA GEMM, 16384 x 2048 x 512:
//   block  = 64 rows x 128 cols, 256 threads (8 waves)
//   wave   = one 16-col n-tile x FOUR 16-row m-subtiles (4 accumulators)
//   A (embedding-gathered rows) staged in LDS once per block per K-chunk:
//     global float4 loads -> ds_store_b128 -> ds_load_b64 fragments
//   B streamed from global (1 b64 per lane per K-step feeds 4 WMMAs)
// Output relaid as [T][B][4H] so each timestep of the recurrence reads one
// contiguous 512 KB block.
// ---------------------------------------------------------------------------
__global__ void xproj_gemm(const int*   __restrict__ x,
                           const float* __restrict__ embed,
                           const float* __restrict__ wi_w,
                           const float* __restrict__ wi_b,
                           float*       __restrict__ xproj) {
    __shared__ float a_s[MBLK][KCHUNK + APAD];

    const int tid   = threadIdx.x;
    const int lane  = tid & 31;
    const int wave  = tid >> 5;
    const int n0    = blockIdx.x * 128 + wave * 16;
    const int m0    = blockIdx.y * MBLK;
    const int khalf = (lane >> 4) * 2;           // lanes 16-31 hold K+2,K+3

    // ---- staging assignment: 4 threads per row, 16 floats each ----
    const int srow = tid >> 2;                   // 0..63
    const int scol = (tid & 3) * 16;             // 0,16,32,48
    const int tok  = x[m0 + srow];               // embed[PAD] row is zero
    const float* asrc = embed + (size_t)tok * E_ + scol;

    // ---- B operand: col N = lane&15 ; B[k][n] = wi_w[n*E + k] ----
    const int    ncol = n0 + (lane & 15);
    const float* bptr = wi_w + (size_t)ncol * E_ + khalf;

    v8f acc0 = {}, acc1 = {}, acc2 = {}, acc3 = {};

    for (int kc = 0; kc < E_; kc += KCHUNK) {
        // stage A chunk (64 x 64 fp32) into LDS
        float4 s0 = *(const float4*)(asrc + kc + 0);
        float4 s1 = *(const float4*)(asrc + kc + 4);
        float4 s2 = *(const float4*)(asrc + kc + 8);
        float4 s3 = *(const float4*)(asrc + kc + 12);
        *(float4*)&a_s[srow][scol + 0]  = s0;    // ds_store_b128
        *(float4*)&a_s[srow][scol + 4]  = s1;
        *(float4*)&a_s[srow][scol + 8]  = s2;
        *(float4*)&a_s[srow][scol + 12] = s3;
        __syncthreads();

        // prefetch next B chunk while consuming this one
        if (kc + KCHUNK < E_)
            __builtin_prefetch(bptr + kc + KCHUNK, 0, 0);

        #pragma unroll 4
        for (int k0 = 0; k0 < KCHUNK; k0 += 4) {
            v2f b  = *(const v2f*)(bptr + kc + k0);                 // global b64
            v2f a0 = *(const v2f*)&a_s[ 0 + (lane & 15)][k0 + khalf]; // ds b64
            v2f a1 = *(const v2f*)&a_s[16 + (lane & 15)][k0 + khalf];
            v2f a2 = *(const v2f*)&a_s[32 + (lane & 15)][k0 + khalf];
            v2f a3 = *(const v2f*)&a_s[48 + (lane & 15)][k0 + khalf];
            acc0 = wmma_f32(a0, b, acc0);
            acc1 = wmma_f32(a1, b, acc1);
            acc2 = wmma_f32(a2, b, acc2);
            acc3 = wmma_f32(a3, b, acc3);
        }
        __syncthreads();
    }

    // ---- epilogue: bias add + scatter into [T][B][4H] ----
    const float bias  = wi_b[ncol];
    const int   rbase = ((lane >> 4) << 3);      // lanes 16-31: rows M=8..15
    v8f accs[4] = {acc0, acc1, acc2, acc3};
    #pragma unroll
    for (int ms = 0; ms < 4; ++ms) {
        #pragma unroll
        for (int r = 0; r < 8; ++r) {
            int m  = m0 + ms * 16 + rbase + r;
            int bb = m >> 8;                     // batch  (T_ == 256)
            int tt = m & (T_ - 1);               // time
            xproj[((size_t)tt * B_ + bb) * G4_ + ncol] = accs[ms][r] + bias;
        }
    }
}

// ---------------------------------------------------------------------------
// One LSTM timestep. One wave owns a 16-batch x 16-hidden patch and carries
// FOUR f32 WMMA accumulators (i,f,g,o gate tiles). The four C/D tiles share
// the same lane<->(M,N) mapping, so the nonlinear cell update is lane-local:
// no LDS, no cross-wave exchange. h is double-buffered (hin -> hout) to kill
// the cross-block RAW hazard on the recurrent matrix operand.
// ---------------------------------------------------------------------------
__global__ void lstm_step(const float* __restrict__ hin,
                          float*       __restrict__ hout,
                          float*       __restrict__ cstate,
                          float*       __restrict__ finalh,
                          const float* __restrict__ xproj,
                          const float* __restrict__ wh_w,
                          const float* __restrict__ wh_b,
                          const int*   __restrict__ x,
                          const int*   __restrict__ len,
                          int t) {
    const int lane  = threadIdx.x & 31;
    const int j0    = blockIdx.x * 16;          // hidden-column tile
    const int m0    = blockIdx.y * 16;          // batch-row tile
    const int khalf = (lane >> 4) * 2;

    const float* aptr = hin + (size_t)(m0 + (lane & 15)) * H_ + khalf;
    const int    jc   = j0 + (lane & 15);

    const float* bp0 = wh_w + (size_t)(0 * H_ + jc) * H_ + khalf;
    const float* bp1 = wh_w + (size_t)(1 * H_ + jc) * H_ + khalf;
    const float* bp2 = wh_w + (size_t)(2 * H_ + jc) * H_ + khalf;
    const float* bp3 = wh_w + (size_t)(3 * H_ + jc) * H_ + khalf;

    v8f ai = {}, af = {}, ag = {}, ao = {};
    #pragma unroll 4
    for (int k0 = 0; k0 < H_; k0 += 4) {
        v2f a  = *(const v2f*)(aptr + k0);
        v2f b0 = *(const v2f*)(bp0 + k0);
        v2f b1 = *(const v2f*)(bp1 + k0);
        v2f b2 = *(const v2f*)(bp2 + k0);
        v2f b3 = *(const v2f*)(bp3 + k0);
        ai = wmma_f32(a, b0, ai);
        af = wmma_f32(a, b1, af);
        ag = wmma_f32(a, b2, ag);
        ao = wmma_f32(a, b3, ao);
    }

    const float bi = wh_b[0 * H_ + jc];
    const float bf = wh_b[1 * H_ + jc];
    const float bg = wh_b[2 * H_ + jc];
    const float bo = wh_b[3 * H_ + jc];

    const int mbase = m0 + ((lane >> 4) << 3);
    #pragma unroll
    for (int r = 0; r < 8; ++r) {
        const int    m  = mbase + r;            // batch row
        const float* xp = xproj + ((size_t)t * B_ + m) * G4_;

        float gi = sigmf(ai[r] + xp[0 * H_ + jc] + bi);
        float gf = sigmf(af[r] + xp[1 * H_ + jc] + bf);
        float gg = tanhf(ag[r] + xp[2 * H_ + jc] + bg);
        float go = sigmf(ao[r] + xp[3 * H_ + jc] + bo);

        const size_t idx = (size_t)m * H_ + jc;
        float cn = gf * cstate[idx] + gi * gg;
        float hn = go * tanhf(cn);
        cstate[idx] = cn;
        hout[idx]   = hn;

        // reference: final = hs[b, len[b]-1], hs zeroed at pad positions
        if (t == len[m] - 1) {
            float msk = (x[m * T_ + t] != PAD_) ? 1.0f : 0.0f;
            finalh[idx] = hn * msk;
        }
    }
}

// ---------------------------------------------------------------------------
// logits[b,o] = finalh[b] . out_w[o] + out_b[o]  (wave32 shuffle reduction)
// ---------------------------------------------------------------------------
__global__ void logits_kernel(const float* __restrict__ finalh,
                              const float* __restrict__ out_w,
                              const float* __restrict__ out_b,
                              float*       __restrict__ out) {
    const int b    = blockIdx.x;
    const int o    = threadIdx.y;
    const int lane = threadIdx.x;

    float s = 0.0f;
    for (int k = lane; k < H_; k += 32)
        s += finalh[(size_t)b * H_ + k] * out_w[(size_t)o * H_ + k];

    #pragma unroll
    for (int off = 16; off > 0; off >>= 1)
        s += __shfl_down(s, off, 32);           // wave32

    if (lane == 0) out[b * O_ + o] = s + out_b[o];
}

// ---------------------------------------------------------------------------
extern "C" void kernel_launch(void* const* d_in, const int* in_sizes, int n_in,
                              void* d_out, int out_size, void* d_ws, size_t ws_size,
                              hipStream_t stream) {
    const int*   x     = (const int*)d_in[0];
    const float* embed = (const float*)d_in[1];
    const float* wi_w  = (const float*)d_in[2];
    const float* wi_b  = (const float*)d_in[3];
    const float* wh_w  = (const float*)d_in[4];
    const float* wh_b  = (const float*)d_in[5];
    const float* out_w = (const float*)d_in[6];
    const float* out_b = (const float*)d_in[7];
    float*       out   = (float*)d_out;

    // workspace layout: xproj (134 MB) | h0 | h1 | c | final | lengths
    char* ws = (char*)d_ws;
    const size_t XPROJ_BYTES = (size_t)T_ * B_ * G4_ * sizeof(float);
    float* xproj = (float*)ws;
    float* h0    = (float*)(ws + XPROJ_BYTES);
    float* h1    = h0 + B_ * H_;
    float* cc    = h1 + B_ * H_;
    float* fin   = cc + B_ * H_;
    int*   len   = (int*)(fin + B_ * H_);

    // zero h0,h1,c,final (contiguous) every call -> deterministic
    hipMemsetAsync(h0, 0, (size_t)4 * B_ * H_ * sizeof(float), stream);
    lengths_kernel<<<1, 64, 0, stream>>>(x, len);

    // 16384x2048x512 fp32 GEMM: 4096 blocks x 256 threads, 64x128 per block
    xproj_gemm<<<dim3(G4_ / 128, (B_ * T_) / MBLK), 256, 0, stream>>>(
        x, embed, wi_w, wi_b, xproj);

    // 256 sequential steps; h double-buffered
    float* hin = h0;
    float* hout = h1;
    for (int t = 0; t < T_; ++t) {
        lstm_step<<<dim3(H_ / 16, B_ / 16), 32, 0, stream>>>(
            hin, hout, cc, fin, xproj, wh_w, wh_b, x, len, t);
        float* tmp = hin; hin = hout; hout = tmp;
    }

    logits_kernel<<<dim3(B_), dim3(32, O_), 0, stream>>>(fin, out_w, out_b, out);
}